// PseudoTokenGridEncoder_83013127897911
// MI455X (gfx1250) — compile-verified
//
#include <hip/hip_runtime.h>
#include <math.h>

#define B 4
#define U 8192
#define E 128
#define GH 64
#define GW 64
#define S (GH * GW)
#define NH 8
#define DH 16
#define MAX_PATCH 24
#define NBUCKET (B * S)      // 16384
#define NPTS (B * U)         // 32768

typedef __attribute__((ext_vector_type(2))) float v2f;
typedef __attribute__((ext_vector_type(8))) float v8f;

// ---------------------------------------------------------------------------
// Kernel 1: bucketize off-grid points into grid cells (atomic slot assignment)
// ---------------------------------------------------------------------------
__global__ __launch_bounds__(256) void bucketize_kernel(
    const float* __restrict__ xc_off, int* __restrict__ counts,
    int* __restrict__ idx_list) {
  int i = blockIdx.x * blockDim.x + threadIdx.x;
  if (i >= NPTS) return;
  int b = i / U;
  float x0 = xc_off[2 * i + 0];
  float x1 = xc_off[2 * i + 1];
  // linspace(0,1,64): step = 1/63 -> index = round(x*63), round-half-even
  int i0 = (int)rintf(x0 * 63.0f);
  i0 = i0 < 0 ? 0 : (i0 > GH - 1 ? GH - 1 : i0);
  int i1 = (int)rintf(x1 * 63.0f);
  i1 = i1 < 0 ? 0 : (i1 > GW - 1 ? GW - 1 : i1);
  int flat = b * S + i0 * GW + i1;
  int slot = atomicAdd(&counts[flat], 1);
  if (slot < MAX_PATCH - 1) idx_list[flat * (MAX_PATCH - 1) + slot] = i;
}

// ---------------------------------------------------------------------------
// Kernel 2: C[M x 128] = A[M x 128] @ W[128 x 128], fp32 WMMA 16x16x4.
// M must be a multiple of 128. 256 threads = 8 waves; wave -> 16x128 stripe.
//
// W is staged in LDS in a k-pair-interleaved layout:
//   W[k][n]  ->  w_lds[(k/2)*WPAIR + 2*n + (k&1)]
// so each WMMA B fragment (K=k0..k0+1 for one column) is ONE contiguous,
// 8B-aligned ds_load_b64 landing directly in an even VGPR pair.
// WPAIR = 288 dwords: 288 mod 64 == 32, so lanes 0-15 (k0,k0+1) and lanes
// 16-31 (k0+2,k0+3) hit disjoint 32-bank halves -> conflict-free.
// ---------------------------------------------------------------------------
#define WPAIR 288
__global__ __launch_bounds__(256) void gemm128_kernel(
    const float* __restrict__ A, const float* __restrict__ W,
    float* __restrict__ C) {
  __shared__ float w_lds[64 * WPAIR];  // 72 KB of the 320 KB WGP pool

  int tid = threadIdx.x;
  // Cooperative permuted fill of W (64 scalar stores per thread)
  for (int i = tid; i < E * E; i += 256) {
    int k = i >> 7;
    int n = i & (E - 1);
    w_lds[(k >> 1) * WPAIR + 2 * n + (k & 1)] = W[i];
  }
  __syncthreads();

  int wave = tid >> 5;
  int lane = tid & 31;
  int lrow = lane & 15;    // M (for A) / N (for B) index within tile
  int lhalf = lane >> 4;   // 0: K=k0,k0+1   1: K=k0+2,k0+3

  int row0 = blockIdx.x * 128 + wave * 16;
  const float* Arow = A + (size_t)(row0 + lrow) * E;

  // Pull the whole A row toward the WGP caches (global_prefetch_b8)
  __builtin_prefetch(Arow, 0, 0);
  __builtin_prefetch(Arow + 64, 0, 0);

  v8f acc[8] = {};  // 8 N-tiles of 16x16 f32

  for (int ks = 0; ks < 32; ++ks) {
    int k0 = ks * 4 + lhalf * 2;                 // even -> 8B aligned
    v2f a = *(const v2f*)(Arow + k0);            // A[row][k0], A[row][k0+1]
    const float* bbase = w_lds + (2 * ks + lhalf) * WPAIR + 2 * lrow;
#pragma unroll
    for (int n = 0; n < 8; ++n) {
      v2f b = *(const v2f*)(bbase + n * 32);     // W[k0][col], W[k0+1][col]
      acc[n] = __builtin_amdgcn_wmma_f32_16x16x4_f32(
          /*neg_a=*/false, a, /*neg_b=*/false, b,
          /*c_mod=*/(short)0, acc[n], /*reuse_a=*/false, /*reuse_b=*/false);
    }
  }

  // C/D layout: VGPR r -> row = r + 8*(lane>=16), col = n*16 + (lane&15)
  int crow0 = row0 + lhalf * 8;
#pragma unroll
  for (int n = 0; n < 8; ++n) {
    int col = n * 16 + lrow;
#pragma unroll
    for (int r = 0; r < 8; ++r) {
      C[(size_t)(crow0 + r) * E + col] = acc[n][r];
    }
  }
}

// ---------------------------------------------------------------------------
// Kernel 3: K/V rows for the fake embedding (1x128 @ 128x128, twice)
// ---------------------------------------------------------------------------
__global__ __launch_bounds__(256) void fake_kv_kernel(
    const float* __restrict__ fake, const float* __restrict__ Wk,
    const float* __restrict__ Wv, float* __restrict__ Kfake,
    float* __restrict__ Vfake) {
  int t = threadIdx.x;  // 256 threads
  const float* Wsel = (t < E) ? Wk : Wv;
  float* out = (t < E) ? Kfake : Vfake;
  int col = t & (E - 1);
  float s = 0.f;
  for (int e = 0; e < E; ++e) s += fake[e] * Wsel[e * E + col];
  out[col] = s;
}

// ---------------------------------------------------------------------------
// Kernel 4: per-cell streaming-softmax attention. One wave32 per cell.
// Lane l owns dims [4l, 4l+4) -> head h = l/4; score reduced over 4 lanes.
// ---------------------------------------------------------------------------
__global__ __launch_bounds__(128) void attn_kernel(
    const int* __restrict__ counts, const int* __restrict__ idx_list,
    const float* __restrict__ Q, const float* __restrict__ Koff,
    const float* __restrict__ Voff, const float* __restrict__ Kon,
    const float* __restrict__ Von, const float* __restrict__ Kfake,
    const float* __restrict__ Vfake, const int* __restrict__ ignore_flag,
    float* __restrict__ Abuf) {
  int lane = threadIdx.x & 31;
  int n = blockIdx.x * 4 + (threadIdx.x >> 5);  // cell id, n = b*S + s
  int s = n & (S - 1);
  int c = counts[n];
  if (c > MAX_PATCH - 1) c = MAX_PATCH - 1;
  int ignore = ignore_flag[0];

  // Sort cell's point indices ascending -> matches reference stable-sort
  // slot order, making summation order (and output bits) deterministic.
  int idx[MAX_PATCH - 1];
  for (int p = 0; p < c; ++p) {
    int j = idx_list[n * (MAX_PATCH - 1) + p];
    int q = p;
    while (q > 0 && idx[q - 1] > j) { idx[q] = idx[q - 1]; --q; }
    idx[q] = j;
  }

  float4 qv = *(const float4*)(Q + (size_t)s * E + lane * 4);

  float m = -INFINITY, lsum = 0.f;
  float4 o = {0.f, 0.f, 0.f, 0.f};

  for (int p = 0; p <= c; ++p) {  // c off-grid tokens + 1 on-grid/fake token
    const float *kr, *vr;
    if (p < c) {
      int j = idx[p];
      kr = Koff + (size_t)j * E;
      vr = Voff + (size_t)j * E;
    } else if (ignore) {
      kr = Kfake;
      vr = Vfake;
    } else {
      kr = Kon + (size_t)n * E;
      vr = Von + (size_t)n * E;
    }
    float4 kv = *(const float4*)(kr + lane * 4);
    float part = qv.x * kv.x + qv.y * kv.y + qv.z * kv.z + qv.w * kv.w;
    part += __shfl_xor(part, 1, 32);
    part += __shfl_xor(part, 2, 32);          // full 16-dim head dot
    float sc = part * 0.25f;                  // 1/sqrt(DH), DH=16

    float mn = fmaxf(m, sc);
    float alpha = __expf(m - mn);             // exp(-inf)=0 on first token
    float w = __expf(sc - mn);
    float4 vv = *(const float4*)(vr + lane * 4);
    o.x = o.x * alpha + w * vv.x;
    o.y = o.y * alpha + w * vv.y;
    o.z = o.z * alpha + w * vv.z;
    o.w = o.w * alpha + w * vv.w;
    lsum = lsum * alpha + w;
    m = mn;
  }

  float inv = 1.0f / lsum;  // lsum > 0: at least the on-grid token is valid
  float4 res = {o.x * inv, o.y * inv, o.z * inv, o.w * inv};
  *(float4*)(Abuf + (size_t)n * E + lane * 4) = res;
}

// ---------------------------------------------------------------------------
// Host launcher
// ---------------------------------------------------------------------------
extern "C" void kernel_launch(void* const* d_in, const int* in_sizes, int n_in,
                              void* d_out, int out_size, void* d_ws,
                              size_t ws_size, hipStream_t stream) {
  const float* xc_off = (const float*)d_in[0];
  // d_in[1] = xc_on_grid: regular linspace grid, reconstructed analytically
  const float* zc_off = (const float*)d_in[2];   // (B,U,E)
  const float* zc_on = (const float*)d_in[3];    // (B,GH,GW,E) == (B*S, E)
  const float* latents = (const float*)d_in[4];  // (S, E)
  const float* fake = (const float*)d_in[5];     // (E,)
  const float* Wq = (const float*)d_in[6];
  const float* Wk = (const float*)d_in[7];
  const float* Wv = (const float*)d_in[8];
  const float* Wo = (const float*)d_in[9];
  const int* ignore_flag = (const int*)d_in[10];

  char* ws = (char*)d_ws;
  size_t off = 0;
  int* counts = (int*)(ws + off);   off += (size_t)NBUCKET * 4;                  // 64 KB
  int* idx_list = (int*)(ws + off); off += (size_t)NBUCKET * (MAX_PATCH - 1) * 4;// 1.44 MB
  float* Qb   = (float*)(ws + off); off += (size_t)S * E * 4;                    // 2 MB
  float* Koff = (float*)(ws + off); off += (size_t)NPTS * E * 4;                 // 16 MB
  float* Voff = (float*)(ws + off); off += (size_t)NPTS * E * 4;                 // 16 MB
  float* Kon  = (float*)(ws + off); off += (size_t)NBUCKET * E * 4;              // 8 MB
  float* Von  = (float*)(ws + off); off += (size_t)NBUCKET * E * 4;              // 8 MB
  float* Abuf = (float*)(ws + off); off += (size_t)NBUCKET * E * 4;              // 8 MB
  float* Kfake = (float*)(ws + off); off += 512;
  float* Vfake = (float*)(ws + off); off += 512;
  (void)ws_size; (void)in_sizes; (void)n_in; (void)out_size;

  // counts must be zero at the start of every call
  hipMemsetAsync(counts, 0, (size_t)NBUCKET * 4, stream);

  bucketize_kernel<<<(NPTS + 255) / 256, 256, 0, stream>>>(xc_off, counts,
                                                           idx_list);

  // Dense fp32 WMMA projections (no padded/zero rows ever materialized)
  gemm128_kernel<<<S / 128, 256, 0, stream>>>(latents, Wq, Qb);        // 4096
  gemm128_kernel<<<NPTS / 128, 256, 0, stream>>>(zc_off, Wk, Koff);    // 32768
  gemm128_kernel<<<NPTS / 128, 256, 0, stream>>>(zc_off, Wv, Voff);
  gemm128_kernel<<<NBUCKET / 128, 256, 0, stream>>>(zc_on, Wk, Kon);   // 16384
  gemm128_kernel<<<NBUCKET / 128, 256, 0, stream>>>(zc_on, Wv, Von);
  fake_kv_kernel<<<1, 256, 0, stream>>>(fake, Wk, Wv, Kfake, Vfake);

  // Per-cell gather + 8-head streaming softmax (1 wave32 per cell)
  attn_kernel<<<NBUCKET / 4, 128, 0, stream>>>(counts, idx_list, Qb, Koff,
                                               Voff, Kon, Von, Kfake, Vfake,
                                               ignore_flag, Abuf);

  // Output projection straight into d_out (B,GH,GW,E) row-major
  gemm128_kernel<<<NBUCKET / 128, 256, 0, stream>>>(Abuf, Wo, (float*)d_out);
}